// Universal_17961553232124
// MI455X (gfx1250) — compile-verified
//
#include <hip/hip_runtime.h>

#define N_NODES 50000
#define N_EDGES 800000
#define FEATS   64
#define CLASSES 16
#define HIDDEN  64
#define DEPTH   10
#define EMB_DIM 7
#define ADJ_HID 11

typedef __attribute__((ext_vector_type(2))) float v2f;
typedef __attribute__((ext_vector_type(8))) float v8f;

// ---------------------------------------------------------------------------
// Graph preprocessing: degree -> inv_sqrt -> edge norm -> CSR (counting sort)
// ---------------------------------------------------------------------------
__global__ void degree_kernel(const int* __restrict__ dst, int* __restrict__ deg, int e) {
    int i = blockIdx.x * blockDim.x + threadIdx.x;
    if (i < e) atomicAdd(&deg[dst[i]], 1);
}

__global__ void invsqrt_kernel(const int* __restrict__ deg, float* __restrict__ inv, int n) {
    int i = blockIdx.x * blockDim.x + threadIdx.x;
    if (i < n) {
        float d = (float)deg[i];
        inv[i] = (d > 0.0f) ? rsqrtf(d) : 0.0f;
    }
}

__global__ void norm_kernel(const int* __restrict__ src, const int* __restrict__ dst,
                            const float* __restrict__ inv, float* __restrict__ norm, int e) {
    int i = blockIdx.x * blockDim.x + threadIdx.x;
    if (i < e) norm[i] = inv[src[i]] * inv[dst[i]];
}

// Single-block hierarchical scan over N (executed once; not perf critical)
__global__ void scan_rowptr_kernel(const int* __restrict__ deg, int* __restrict__ row_ptr, int n) {
    __shared__ int sh[1024];
    __shared__ int carry;
    int t = threadIdx.x;
    if (t == 0) { carry = 0; row_ptr[0] = 0; }
    __syncthreads();
    for (int base = 0; base < n; base += 1024) {
        int v = (base + t < n) ? deg[base + t] : 0;
        sh[t] = v;
        __syncthreads();
        for (int off = 1; off < 1024; off <<= 1) {
            int add = (t >= off) ? sh[t - off] : 0;
            __syncthreads();
            sh[t] += add;
            __syncthreads();
        }
        if (base + t < n) row_ptr[base + t + 1] = sh[t] + carry;
        __syncthreads();
        if (t == 1023) carry += sh[1023];
        __syncthreads();
    }
}

__global__ void csr_fill_kernel(const int* __restrict__ src, const int* __restrict__ dst,
                                const float* __restrict__ norm,
                                const int* __restrict__ row_ptr, int* __restrict__ cursor,
                                int* __restrict__ csr_src, float* __restrict__ csr_w, int e) {
    int i = blockIdx.x * blockDim.x + threadIdx.x;
    if (i < e) {
        int d = dst[i];
        int slot = row_ptr[d] + atomicAdd(&cursor[d], 1);
        csr_src[slot] = src[i];
        csr_w[slot]   = norm[i];
    }
}

// ---------------------------------------------------------------------------
// Diffusion round: zout = 0.9 * (D^-1/2 A D^-1/2) zin + 0.1 * h0
// Pure gather (no atomics), vectorized x4: each thread owns 4 contiguous
// feats, so each neighbor row fetch is a single global_load_b128 and only
// F/4 threads per node walk the (same-address, wave-broadcast) CSR lists.
// CSR + zin resident in 192MB L2 -> L2-request/issue bound, minimize ops.
// ---------------------------------------------------------------------------
template <int FQ>  // FQ = F/4 (16 for feats, 4 for classes)
__global__ void conv_kernel(const int* __restrict__ row_ptr, const int* __restrict__ csr_src,
                            const float* __restrict__ csr_w, const float4* __restrict__ zin,
                            const float4* __restrict__ h0, float4* __restrict__ zout, int n) {
    int idx  = blockIdx.x * blockDim.x + threadIdx.x;
    int node = idx / FQ;
    int fq   = idx - node * FQ;
    if (node >= n) return;
    int s0 = row_ptr[node];
    int s1 = row_ptr[node + 1];
    float4 acc = make_float4(0.f, 0.f, 0.f, 0.f);
    for (int s = s0; s < s1; ++s) {
        int   sn = csr_src[s];
        float w  = csr_w[s];
        if (s + 1 < s1) __builtin_prefetch(&zin[csr_src[s + 1] * FQ], 0, 1);
        float4 v = zin[sn * FQ + fq];
        acc.x = fmaf(w, v.x, acc.x);
        acc.y = fmaf(w, v.y, acc.y);
        acc.z = fmaf(w, v.z, acc.z);
        acc.w = fmaf(w, v.w, acc.w);
    }
    float4 hv = h0[idx];
    float4 o;
    o.x = fmaf(acc.x, 0.9f, 0.1f * hv.x);
    o.y = fmaf(acc.y, 0.9f, 0.1f * hv.y);
    o.z = fmaf(acc.z, 0.9f, 0.1f * hv.z);
    o.w = fmaf(acc.w, 0.9f, 0.1f * hv.w);
    zout[idx] = o;
}

// ---------------------------------------------------------------------------
// Per-(node,feat) pair MLP: in = [z, x, emb[f]] (9) -> relu(@A1[9,11]) -> @A2[11,1] / 2
// ---------------------------------------------------------------------------
__global__ void pair_mlp_kernel(const float* __restrict__ z, const float* __restrict__ x,
                                const float* __restrict__ A1, const float* __restrict__ ab1,
                                const float* __restrict__ A2, const float* __restrict__ ab2,
                                const float* __restrict__ emb, float* __restrict__ zp, int n) {
    int idx = blockIdx.x * blockDim.x + threadIdx.x;
    if (idx >= n * FEATS) return;
    int f = idx & (FEATS - 1);
    float in[2 + EMB_DIM];
    in[0] = z[idx];
    in[1] = x[idx];
#pragma unroll
    for (int i = 0; i < EMB_DIM; ++i) in[2 + i] = emb[f * EMB_DIM + i];
    float out = ab2[0];
#pragma unroll
    for (int j = 0; j < ADJ_HID; ++j) {
        float u = ab1[j];
#pragma unroll
        for (int i = 0; i < 2 + EMB_DIM; ++i) u = fmaf(in[i], A1[i * ADJ_HID + j], u);
        u = fmaxf(u, 0.0f);
        out = fmaf(u, A2[j], out);
    }
    zp[idx] = out * 0.5f;
}

// ---------------------------------------------------------------------------
// GEMM1: H[N,64] = relu( concat(zp, x)[N,128] @ W1[128,64] + b1 )
// One wave per 16x16 output tile, f32 WMMA (16x16x4) keeps exact f32 precision.
// A (16x4 f32): VGPR0 = K0 (lanes 0-15) / K2 (lanes 16-31); VGPR1 = K1 / K3.
// B (4x16 f32): VGPR0 = row K0 (lanes 0-15) / row K2 (lanes 16-31); VGPR1 = K1 / K3.
// C (16x16 f32): c[r] = C[r + 8*half][lane&15].
// ---------------------------------------------------------------------------
__global__ void gemm1_kernel(const float* __restrict__ zp, const float* __restrict__ x,
                             const float* __restrict__ W1, const float* __restrict__ b1,
                             float* __restrict__ h, int n) {
    int wave  = (blockIdx.x * blockDim.x + threadIdx.x) >> 5;
    int lane  = threadIdx.x & 31;
    int tiles = (n >> 4) * 4;                 // row-tiles x 4 col-tiles of 16
    if (wave >= tiles) return;                // wave-uniform: EXEC stays all-1s for WMMA
    int tr = wave >> 2, tc = wave & 3;
    int half = lane >> 4, lr = lane & 15;
    int col = tc * 16 + lr;
    int m   = tr * 16 + lr;

    v8f c;
    float bv = b1[col];
#pragma unroll
    for (int r = 0; r < 8; ++r) c[r] = bv;

#pragma unroll
    for (int kt = 0; kt < 32; ++kt) {
        int ka = kt * 4 + 2 * half;           // K index this half-wave supplies
        const float* abase = (ka < FEATS) ? (zp + m * FEATS + ka)
                                          : (x + m * FEATS + (ka - FEATS));
        v2f a, b;
        a.x = abase[0];
        a.y = abase[1];
        b.x = W1[ka * HIDDEN + col];
        b.y = W1[(ka + 1) * HIDDEN + col];
        c = __builtin_amdgcn_wmma_f32_16x16x4_f32(false, a, false, b, (short)0, c, false, false);
    }
#pragma unroll
    for (int r = 0; r < 8; ++r) {
        int row = tr * 16 + half * 8 + r;
        h[row * HIDDEN + col] = fmaxf(c[r], 0.0f);   // relu epilogue
    }
}

// GEMM2: logits[N,16] = H[N,64] @ W2[64,16] + b2   (one wave per 16-row tile)
__global__ void gemm2_kernel(const float* __restrict__ h, const float* __restrict__ W2,
                             const float* __restrict__ b2, float* __restrict__ out, int n) {
    int wave  = (blockIdx.x * blockDim.x + threadIdx.x) >> 5;
    int lane  = threadIdx.x & 31;
    int tiles = n >> 4;
    if (wave >= tiles) return;
    int half = lane >> 4, lr = lane & 15;
    int m = wave * 16 + lr;

    v8f c;
    float bv = b2[lr];
#pragma unroll
    for (int r = 0; r < 8; ++r) c[r] = bv;

#pragma unroll
    for (int kt = 0; kt < 16; ++kt) {
        int ka = kt * 4 + 2 * half;
        v2f a, b;
        a.x = h[m * HIDDEN + ka];
        a.y = h[m * HIDDEN + ka + 1];
        b.x = W2[ka * CLASSES + lr];
        b.y = W2[(ka + 1) * CLASSES + lr];
        c = __builtin_amdgcn_wmma_f32_16x16x4_f32(false, a, false, b, (short)0, c, false, false);
    }
#pragma unroll
    for (int r = 0; r < 8; ++r) {
        int row = wave * 16 + half * 8 + r;
        out[row * CLASSES + lr] = c[r];
    }
}

// ---------------------------------------------------------------------------
// Host orchestration
// ---------------------------------------------------------------------------
static inline size_t align256(size_t v) { return (v + 255) & ~(size_t)255; }

extern "C" void kernel_launch(void* const* d_in, const int* in_sizes, int n_in,
                              void* d_out, int out_size, void* d_ws, size_t ws_size,
                              hipStream_t stream) {
    const float* x   = (const float*)d_in[0];
    const int*   src = (const int*)  d_in[1];
    const int*   dst = (const int*)  d_in[2];
    const float* W1  = (const float*)d_in[3];
    const float* b1  = (const float*)d_in[4];
    const float* W2  = (const float*)d_in[5];
    const float* b2  = (const float*)d_in[6];
    const float* A1  = (const float*)d_in[7];
    const float* ab1 = (const float*)d_in[8];
    const float* A2  = (const float*)d_in[9];
    const float* ab2 = (const float*)d_in[10];
    const float* emb = (const float*)d_in[11];
    float* out = (float*)d_out;

    const int n = N_NODES, e = N_EDGES;

    // Workspace layout
    char*  base = (char*)d_ws;
    size_t off  = 0;
    int*   deg      = (int*)  (base + off); off += align256(sizeof(int)   * n);
    float* inv      = (float*)(base + off); off += align256(sizeof(float) * n);
    float* norm     = (float*)(base + off); off += align256(sizeof(float) * e);
    int*   row_ptr  = (int*)  (base + off); off += align256(sizeof(int)   * (n + 1));
    int*   cursor   = (int*)  (base + off); off += align256(sizeof(int)   * n);
    int*   csr_src  = (int*)  (base + off); off += align256(sizeof(int)   * e);
    float* csr_w    = (float*)(base + off); off += align256(sizeof(float) * e);
    float* zA       = (float*)(base + off); off += align256(sizeof(float) * n * FEATS);
    float* zB       = (float*)(base + off); off += align256(sizeof(float) * n * FEATS);
    float* zp       = (float*)(base + off); off += align256(sizeof(float) * n * FEATS);
    float* hbuf     = (float*)(base + off); off += align256(sizeof(float) * n * HIDDEN);
    float* logit0   = (float*)(base + off); off += align256(sizeof(float) * n * CLASSES);
    float* logitA   = (float*)(base + off); off += align256(sizeof(float) * n * CLASSES);
    float* logitB   = (float*)(base + off); off += align256(sizeof(float) * n * CLASSES);
    (void)off; (void)ws_size; (void)in_sizes; (void)n_in; (void)out_size;

    const int TPB = 256;
    const int EB  = (e + TPB - 1) / TPB;
    const int NB  = (n + TPB - 1) / TPB;

    // --- CSR build (one-time) ---
    hipMemsetAsync(deg,    0, sizeof(int) * n, stream);
    hipMemsetAsync(cursor, 0, sizeof(int) * n, stream);
    degree_kernel  <<<EB, TPB, 0, stream>>>(dst, deg, e);
    invsqrt_kernel <<<NB, TPB, 0, stream>>>(deg, inv, n);
    norm_kernel    <<<EB, TPB, 0, stream>>>(src, dst, inv, norm, e);
    scan_rowptr_kernel<<<1, 1024, 0, stream>>>(deg, row_ptr, n);
    csr_fill_kernel<<<EB, TPB, 0, stream>>>(src, dst, norm, row_ptr, cursor, csr_src, csr_w, e);

    // --- Diffusion loop 1 (feature width 64, vec4 -> 16 lanes/node) ---
    const int C64B = (n * (FEATS / 4) + TPB - 1) / TPB;
    const float4* zin64 = (const float4*)x;
    for (int i = 0; i < DEPTH; ++i) {
        float4* zout = (float4*)((i & 1) ? zB : zA);
        conv_kernel<FEATS / 4><<<C64B, TPB, 0, stream>>>(row_ptr, csr_src, csr_w,
                                                         zin64, (const float4*)x, zout, n);
        zin64 = zout;
    }
    const float* zfin = (const float*)zin64;   // zB for DEPTH=10

    // --- Pair MLP ---
    const int PB = (n * FEATS + TPB - 1) / TPB;
    pair_mlp_kernel<<<PB, TPB, 0, stream>>>(zfin, x, A1, ab1, A2, ab2, emb, zp, n);

    // --- Dense node MLP via f32 WMMA ---
    const int tiles1 = (n >> 4) * 4;                 // 12500 tiles, 8 waves/block
    gemm1_kernel<<<(tiles1 + 7) / 8, TPB, 0, stream>>>(zp, x, W1, b1, hbuf, n);
    const int tiles2 = n >> 4;                       // 3125 tiles
    gemm2_kernel<<<(tiles2 + 7) / 8, TPB, 0, stream>>>(hbuf, W2, b2, logit0, n);

    // --- Diffusion loop 2 (class width 16, vec4 -> 4 lanes/node) ---
    const int C16B = (n * (CLASSES / 4) + TPB - 1) / TPB;
    const float4* zin16 = (const float4*)logit0;
    for (int i = 0; i < DEPTH; ++i) {
        float4* zout = (i == DEPTH - 1) ? (float4*)out
                                        : (float4*)((i & 1) ? logitB : logitA);
        conv_kernel<CLASSES / 4><<<C16B, TPB, 0, stream>>>(row_ptr, csr_src, csr_w,
                                                           zin16, (const float4*)logit0, zout, n);
        zin16 = zout;
    }
}